// Re_Graph_51402168599351
// MI455X (gfx1250) — compile-verified
//
#include <hip/hip_runtime.h>
#include <hip/hip_bf16.h>

// Problem constants from the reference
#define B_ 64
#define C_ 1024
#define H_ 16
#define F_ 256           // H*H
#define ROWS_ (B_ * C_)  // 65536
#define KMAX_ 16

typedef __attribute__((ext_vector_type(2))) float v2f;
typedef __attribute__((ext_vector_type(8))) float v8f;

// ---------------------------------------------------------------------------
// Kernel 1: GAP — one wave (32 lanes) reduces one 256-float channel row.
// ---------------------------------------------------------------------------
__global__ void gap_kernel(const float* __restrict__ x, float* __restrict__ gap) {
  int row  = blockIdx.x * 8 + (threadIdx.x >> 5);  // 8 waves / block
  int lane = threadIdx.x & 31;
  const float4* p = (const float4*)(x + (size_t)row * F_);
  float4 a = p[lane * 2];
  float4 b = p[lane * 2 + 1];
  float s = a.x + a.y + a.z + a.w + b.x + b.y + b.z + b.w;
#pragma unroll
  for (int off = 16; off >= 1; off >>= 1) s += __shfl_xor(s, off, 32);
  if (lane == 0) gap[row] = s * (1.0f / (float)F_);
}

// ---------------------------------------------------------------------------
// Kernel 2: top-k neighbors per node -> symmetric adjacency bitmask (+self).
// One block = 256 nodes of one batch quarter; gap row cached in LDS.
// ---------------------------------------------------------------------------
__global__ void topk_adj_kernel(const float* __restrict__ gap,
                                unsigned* __restrict__ Abits,
                                const int* __restrict__ kptr) {
  __shared__ float g[C_];
  int b = blockIdx.x >> 2;
  int i = ((blockIdx.x & 3) << 8) + threadIdx.x;
  int k = *kptr;
  if (k > KMAX_) k = KMAX_;
  for (int t = threadIdx.x; t < C_; t += 256) g[t] = gap[(size_t)b * C_ + t];
  __syncthreads();

  float gi = g[i];
  float bd[KMAX_];
  int   bi[KMAX_];
#pragma unroll
  for (int n = 0; n < KMAX_; n++) { bd[n] = 3.4e38f; bi[n] = 0; }

  for (int j = 0; j < C_; j++) {
    if (j == i) continue;
    float d = gi - g[j];
    d = d * d;
    if (d < bd[k - 1]) {            // strict <: ties keep earlier j (top_k order)
      int p = k - 1;
      while (p > 0 && bd[p - 1] > d) {
        bd[p] = bd[p - 1]; bi[p] = bi[p - 1]; --p;
      }
      bd[p] = d; bi[p] = j;
    }
  }

  unsigned* rowb = Abits + ((size_t)b * C_ + i) * (C_ / 32);
  atomicOr(&rowb[i >> 5], 1u << (i & 31));  // self loop (A + I)
  for (int n = 0; n < k; n++) {
    int j = bi[n];
    atomicOr(&rowb[j >> 5], 1u << (j & 31));                              // A[i,j]
    atomicOr(&Abits[((size_t)b * C_ + j) * (C_ / 32) + (i >> 5)],         // A[j,i]
             1u << (i & 31));
  }
}

// ---------------------------------------------------------------------------
// Kernel 3: dinv = rsqrt(degree) from bitmask popcount.
// ---------------------------------------------------------------------------
__global__ void dinv_kernel(const unsigned* __restrict__ Abits,
                            float* __restrict__ dinv) {
  int r = blockIdx.x * 256 + threadIdx.x;
  const unsigned* p = Abits + (size_t)r * (C_ / 32);
  int s = 0;
#pragma unroll
  for (int w = 0; w < C_ / 32; w++) s += __popc(p[w]);
  dinv[r] = rsqrtf((float)s);
}

// ---------------------------------------------------------------------------
// Kernel 4: XW = X @ W via V_WMMA_F32_16X16X4_F32.
// One wave computes one 16x16 tile; K=256 in steps of 4.
// A frag (16x4 f32): lane l<16 -> A[M=l][K..K+1], lanes 16-31 -> K+2..K+3.
// B frag (4x16 f32): VGPR v, lane group h -> row K = k + 2*h + v, N = lane&15.
// C/D: VGPR r -> M = r + 8*(lane>>4), N = lane&15.
// ---------------------------------------------------------------------------
__global__ void gemm_wmma_kernel(const float* __restrict__ X,
                                 const float* __restrict__ W,
                                 float* __restrict__ XW) {
  int wave = (int)((blockIdx.x * blockDim.x + threadIdx.x) >> 5);
  int lane = threadIdx.x & 31;
  int mt = wave >> 4;          // 4096 M-tiles
  int nt = wave & 15;          // 16 N-tiles
  int m0 = mt * 16, n0 = nt * 16;
  int half = lane >> 4, l = lane & 15;

  const float* arow = X + (size_t)(m0 + l) * F_ + 2 * half;     // A[M=l][k+2h]
  const float* bcol = W + (size_t)(2 * half) * F_ + n0 + l;     // W[k+2h][n0+l]

  v8f acc = {};
#pragma unroll 4
  for (int k = 0; k < F_; k += 4) {
    float2 av = *(const float2*)(arow + k);                     // K, K+1 (8B aligned)
    v2f a = {av.x, av.y};
    v2f bb = {bcol[(size_t)k * F_], bcol[(size_t)k * F_ + F_]}; // rows K, K+1
    acc = __builtin_amdgcn_wmma_f32_16x16x4_f32(
        /*neg_a=*/false, a, /*neg_b=*/false, bb,
        /*c_mod=*/(short)0, acc, /*reuse_a=*/false, /*reuse_b=*/false);
  }

  float* out = XW + (size_t)m0 * F_ + n0 + l;
#pragma unroll
  for (int r = 0; r < 8; r++)
    out[(size_t)(r + 8 * half) * F_] = acc[r];
}

// ---------------------------------------------------------------------------
// Kernel 5: out = relu(dinv_i * sum_j dinv_j * XW[j,:] + bias) + fmap
// One block per (b,i); bitmask -> LDS neighbor list with pre-scaled weights.
// ---------------------------------------------------------------------------
__global__ void aggregate_kernel(const float* __restrict__ XW,
                                 const unsigned* __restrict__ Abits,
                                 const float* __restrict__ dinv,
                                 const float* __restrict__ bias,
                                 const float* __restrict__ fmap,
                                 float* __restrict__ out) {
  __shared__ int   nbrj[C_];
  __shared__ float nbrw[C_];
  __shared__ int   cnt;
  int b = blockIdx.x >> 10;
  int i = blockIdx.x & (C_ - 1);
  size_t row = (size_t)b * C_ + i;

  if (threadIdx.x == 0) cnt = 0;
  __syncthreads();
  if (threadIdx.x < C_ / 32) {
    unsigned w = Abits[row * (C_ / 32) + threadIdx.x];
    while (w) {
      int bit = __ffs((int)w) - 1;
      w &= w - 1;
      int j = (int)threadIdx.x * 32 + bit;
      int p = atomicAdd(&cnt, 1);
      nbrj[p] = j;
      nbrw[p] = dinv[(size_t)b * C_ + j];
    }
  }
  __syncthreads();

  int n = cnt;
  float di = dinv[row];
  int f = threadIdx.x;
  float acc = 0.0f;
  for (int t = 0; t < n; t++)
    acc += nbrw[t] * XW[((size_t)b * C_ + nbrj[t]) * F_ + f];
  float v = fmaxf(di * acc + bias[f], 0.0f);
  out[row * F_ + f] = v + fmap[row * F_ + f];
}

// ---------------------------------------------------------------------------
extern "C" void kernel_launch(void* const* d_in, const int* in_sizes, int n_in,
                              void* d_out, int out_size, void* d_ws, size_t ws_size,
                              hipStream_t stream) {
  const float* fmap = (const float*)d_in[0];  // [64,1024,16,16] f32
  const float* W    = (const float*)d_in[1];  // [256,256] f32
  const float* bias = (const float*)d_in[2];  // [256] f32
  const int*   kptr = (const int*)d_in[3];    // scalar k
  float* out = (float*)d_out;

  // Workspace layout (bytes)
  char* ws = (char*)d_ws;
  float*    gap   = (float*)(ws);                                   // 256 KB
  float*    dinv  = (float*)(ws + (size_t)ROWS_ * 4);               // 256 KB
  unsigned* Abits = (unsigned*)(ws + (size_t)ROWS_ * 8);            // 8 MB
  float*    XW    = (float*)(ws + (size_t)ROWS_ * 8 +
                             (size_t)ROWS_ * (C_ / 32) * 4);        // 64 MB

  // 1) GAP: 65536 rows, 8 waves/block
  gap_kernel<<<ROWS_ / 8, 256, 0, stream>>>(fmap, gap);

  // 2) zero adjacency bitmask, then top-k + symmetrize + self loops
  hipMemsetAsync(Abits, 0, (size_t)ROWS_ * (C_ / 32) * 4, stream);
  topk_adj_kernel<<<B_ * 4, 256, 0, stream>>>(gap, Abits, kptr);

  // 3) dinv = rsqrt(degree)
  dinv_kernel<<<ROWS_ / 256, 256, 0, stream>>>(Abits, dinv);

  // 4) XW = X @ W  (WMMA f32 16x16x4); 65536 tiles, 8 waves/block
  gemm_wmma_kernel<<<ROWS_ / 8, 256, 0, stream>>>(fmap, W, XW);

  // 5) sparse normalized aggregation + bias + relu + residual
  aggregate_kernel<<<ROWS_, 256, 0, stream>>>(XW, Abits, dinv, bias, fmap, out);
}